// GCMCLayer_84335977824413
// MI455X (gfx1250) — compile-verified
//
#include <hip/hip_runtime.h>

#define R_REL   5
#define E_EDGE  500000
#define N_USER_ 100000
#define N_NODE  200000
#define D_FEAT  64

typedef float v2f __attribute__((ext_vector_type(2)));
typedef float v8f __attribute__((ext_vector_type(8)));

// ---------------------------------------------------------------------------
// Zero the N x D accumulator in workspace (d_ws is poisoned, not zeroed).
// ---------------------------------------------------------------------------
__global__ void gcmc_zero_kernel(float* __restrict__ p, long long n) {
  long long i = (long long)blockIdx.x * blockDim.x + threadIdx.x;
  const long long stride = (long long)gridDim.x * blockDim.x;
  for (; i < n; i += stride) p[i] = 0.0f;
}

// ---------------------------------------------------------------------------
// Per-relation edge kernel:
//   rf   = review_feat[r] @ review_w[r].T          (WMMA f32 16x16x4, K=64)
//   m    = (x[src] + conv_weight[r][src] + rf) * ci[src]
//   feat[dst] += m                                  (global_atomic_add_f32)
// One wave handles a 16-edge x 64-feature tile (4 C-tiles of 16x16).
// Block = 256 threads = 8 waves = 128 edges. Grid = (E/128 tiles, R).
// ---------------------------------------------------------------------------
__global__ __launch_bounds__(256) void gcmc_edge_kernel(
    const float* __restrict__ x,    // [N, D]
    const float* __restrict__ ci,   // [N]
    const float* __restrict__ rf,   // [R, E, D]
    const int*   __restrict__ src,  // [R, E]
    const int*   __restrict__ dst,  // [R, E]
    const float* __restrict__ cw,   // [R, N, D]
    const float* __restrict__ rw,   // [R, D, D]  (Linear weight, [out,in])
    float*       __restrict__ feat) // [N, D] accumulator
{
  __shared__ float W[D_FEAT * D_FEAT];            // review_w[r], 16 KB
  const int r = blockIdx.y;

  // Cooperative stage of review_w[r] into LDS (4096 floats / 256 threads).
  {
    const float4* ws = (const float4*)(rw + (size_t)r * D_FEAT * D_FEAT);
    float4* wd = (float4*)W;
#pragma unroll
    for (int i = 0; i < 4; ++i)
      wd[threadIdx.x + i * 256] = ws[threadIdx.x + i * 256];
  }
  __syncthreads();

  const int wave = threadIdx.x >> 5;
  const int lane = threadIdx.x & 31;
  const int tile = blockIdx.x * 8 + wave;         // 16 edges per tile
  if (tile * 16 >= E_EDGE) return;
  const long long ebase = (long long)tile * 16;

  const int row = lane & 15;   // A: M row within tile; B/C: N column
  const int hi  = lane >> 4;   // 0 -> K%4 in {0,1}, 1 -> K%4 in {2,3}

  // A fragments: 16 x 64 f32 edge tile of review_feat, per the 16x4 layout.
  // Lane holds edge row (ebase+row), K pair (4*kt + 2*hi, +1): strided b64s.
  const float* rfp = rf + ((size_t)r * E_EDGE + ebase + row) * D_FEAT + hi * 2;
  v2f a[16];
#pragma unroll
  for (int kt = 0; kt < 16; ++kt)
    a[kt] = *(const v2f*)(rfp + kt * 4);

  // 4 N-tiles (64 feature columns), K-loop of 16 WMMAs each; A reused.
  v8f acc[4] = {v8f{}, v8f{}, v8f{}, v8f{}};
#pragma unroll
  for (int nt = 0; nt < 4; ++nt) {
#pragma unroll
    for (int kt = 0; kt < 16; ++kt) {
      // B[k][n] = W[nt*16+n][4*kt+k]; lane's pair from LDS (ds_load_b64).
      v2f b = *(const v2f*)(&W[(nt * 16 + row) * D_FEAT + kt * 4 + hi * 2]);
      acc[nt] = __builtin_amdgcn_wmma_f32_16x16x4_f32(
          false, a[kt], false, b, (short)0, acc[nt], false, false);
    }
  }

  // Epilogue. C layout: VGPR i -> row (hi*8 + i), column nt*16 + row.
  int   sidx[8], didx[8];
  float cis[8];
#pragma unroll
  for (int i = 0; i < 8; ++i) {
    const long long e = (long long)r * E_EDGE + ebase + hi * 8 + i;
    sidx[i] = src[e];
    didx[i] = dst[e];
    cis[i]  = ci[sidx[i]];
  }
#pragma unroll
  for (int nt = 0; nt < 4; ++nt) {
    const int f = nt * 16 + row;
#pragma unroll
    for (int i = 0; i < 8; ++i) {
      const int s = sidx[i];
      const float v =
          (x[(size_t)s * D_FEAT + f] +
           cw[((size_t)r * N_NODE + s) * D_FEAT + f] +
           acc[nt][i]) * cis[i];
      unsafeAtomicAdd(&feat[(size_t)didx[i] * D_FEAT + f], v);
    }
  }
}

// ---------------------------------------------------------------------------
// Final dense layer: out[n] = (feat[n] * ci[n]) @ Wsel.T + bsel,
// Wsel/bsel = fc_user for n < N_USER, fc_item otherwise.
// N_USER is a multiple of 16 so 16-row tiles never straddle the boundary.
// ---------------------------------------------------------------------------
__global__ __launch_bounds__(256) void gcmc_out_kernel(
    const float* __restrict__ feat, // [N, D]
    const float* __restrict__ ci,   // [N]
    const float* __restrict__ wu,   // [D, D]
    const float* __restrict__ bu,   // [D]
    const float* __restrict__ wi,   // [D, D]
    const float* __restrict__ bi,   // [D]
    float*       __restrict__ out)  // [N, D]
{
  __shared__ float W[2 * D_FEAT * D_FEAT];        // both weights, 32 KB
  {
    const float4* su = (const float4*)wu;
    const float4* si = (const float4*)wi;
    float4* wd = (float4*)W;
#pragma unroll
    for (int i = 0; i < 4; ++i) {
      wd[threadIdx.x + i * 256]        = su[threadIdx.x + i * 256];
      wd[1024 + threadIdx.x + i * 256] = si[threadIdx.x + i * 256];
    }
  }
  __syncthreads();

  const int wave = threadIdx.x >> 5;
  const int lane = threadIdx.x & 31;
  const int tile = blockIdx.x * 8 + wave;         // 16 node rows per tile
  if (tile * 16 >= N_NODE) return;
  const int nbase = tile * 16;

  const int row = lane & 15;
  const int hi  = lane >> 4;

  const bool is_item = (nbase >= N_USER_);
  const float* Wl   = W + (is_item ? D_FEAT * D_FEAT : 0);
  const float* bias = is_item ? bi : bu;

  // A fragments: 16 x 64 tile of feat, with the dst ci norm folded in.
  const float cin = ci[nbase + row];
  const float* fp = feat + (size_t)(nbase + row) * D_FEAT + hi * 2;
  v2f a[16];
#pragma unroll
  for (int kt = 0; kt < 16; ++kt) {
    v2f av = *(const v2f*)(fp + kt * 4);
    a[kt]  = av * cin;
  }

  v8f acc[4] = {v8f{}, v8f{}, v8f{}, v8f{}};
#pragma unroll
  for (int nt = 0; nt < 4; ++nt) {
#pragma unroll
    for (int kt = 0; kt < 16; ++kt) {
      v2f b = *(const v2f*)(&Wl[(nt * 16 + row) * D_FEAT + kt * 4 + hi * 2]);
      acc[nt] = __builtin_amdgcn_wmma_f32_16x16x4_f32(
          false, a[kt], false, b, (short)0, acc[nt], false, false);
    }
  }

#pragma unroll
  for (int nt = 0; nt < 4; ++nt) {
    const int f = nt * 16 + row;
    const float bv = bias[f];
#pragma unroll
    for (int i = 0; i < 8; ++i) {
      const int n = nbase + hi * 8 + i;
      out[(size_t)n * D_FEAT + f] = acc[nt][i] + bv;
    }
  }
}

// ---------------------------------------------------------------------------
extern "C" void kernel_launch(void* const* d_in, const int* in_sizes, int n_in,
                              void* d_out, int out_size, void* d_ws, size_t ws_size,
                              hipStream_t stream) {
  const float* x   = (const float*)d_in[0];
  const float* ci  = (const float*)d_in[1];
  const float* rf  = (const float*)d_in[2];
  const int*   src = (const int*)d_in[3];
  const int*   dst = (const int*)d_in[4];
  const float* cw  = (const float*)d_in[5];
  const float* rw  = (const float*)d_in[6];
  const float* wu  = (const float*)d_in[7];
  const float* bu  = (const float*)d_in[8];
  const float* wi  = (const float*)d_in[9];
  const float* bi  = (const float*)d_in[10];
  float* out  = (float*)d_out;
  float* feat = (float*)d_ws;   // N*D floats = 51.2 MB accumulator

  gcmc_zero_kernel<<<1024, 256, 0, stream>>>(feat, (long long)N_NODE * D_FEAT);

  dim3 g1((E_EDGE / 16 + 7) / 8, R_REL);   // 3907 x 5 blocks
  gcmc_edge_kernel<<<g1, 256, 0, stream>>>(x, ci, rf, src, dst, cw, rw, feat);

  const int g2 = (N_NODE / 16 + 7) / 8;    // 1563 blocks
  gcmc_out_kernel<<<g2, 256, 0, stream>>>(feat, ci, wu, bu, wi, bi, out);
}